// ImageSmoother_33698313404615
// MI455X (gfx1250) — compile-verified
//
#include <hip/hip_runtime.h>

// ---------------------------------------------------------------------------
// 7x7 median pool (reflect pad) + hardtanh[0,1] on (8,3,512,512) fp32.
//
// MI455X plan:
//  - HBM floor ~2.2us @ 23.3 TB/s; kernel is VALU-bound on the median
//    selection network -> minimize compare-exchanges (CEs).
//  - Forgetful selection with 2x2 QUAD sharing: the four windows of a 2x2
//    output block share a 6x6 (36-value) core. Shared prefix: seed 26 +
//    stream 10 shared values (buffer 26->16), run ONCE. Fork per pixel and
//    stream its private 13 values (edge row of 7 + edge col of 6; 16->3,
//    then med3). ~251 CEs (~503 v_min/v_max) per pixel vs 480 unshared.
//    Exactness: at round R the ejected min has (25-R)+R = 25 values known
//    >= it, so it cannot be rank 25 of 49 — independent of stream order,
//    which is what legalizes the shared prefix.
//  - 64x16 output tile / 256-thread (8-wave, wave32) workgroup; one 2x2
//    quad per thread. 70x22 halo tile in LDS (6.2 KB), reflect padding
//    resolved once at fill time.
//  - Tile fill uses gfx1250 GLOBAL_LOAD_ASYNC_TO_LDS_B32 (ASYNCcnt +
//    s_wait_asynccnt) when available; plain load+ds_store otherwise.
//  - Lane LDS stride is 2 dwords over 64 banks -> conflict-free.
//  - Outputs stored as float2 (global_store_b64), 8B-aligned.
// ---------------------------------------------------------------------------

#define KWIN 7
#define PADR 3
#define IMG_H 512
#define IMG_W 512
#define TILE_W 64
#define TILE_H 16
#define PITCH (TILE_W + 2 * PADR) /* 70 */
#define TROWS (TILE_H + 2 * PADR) /* 22 */
#define NTHREADS 256

#if defined(__gfx1250__) && __has_builtin(__builtin_amdgcn_global_load_async_to_lds_b32) && \
    __has_builtin(__builtin_amdgcn_s_wait_asynccnt)
#define USE_ASYNC_LDS 1
typedef __attribute__((address_space(1))) int g_int_t;  // AS1 global int *
typedef __attribute__((address_space(3))) int l_int_t;  // AS3 LDS int *
#else
#define USE_ASYNC_LDS 0
#endif

__device__ __forceinline__ int reflect_idx(int i, int n) {
  i = (i < 0) ? -i : i;            // mirror without edge duplication
  i = (i >= n) ? (2 * n - 2 - i) : i;
  return i;
}

// compare-exchange: x <- min, y <- max  (v_min_num_f32 / v_max_num_f32)
__device__ __forceinline__ void ce(float& x, float& y) {
  float lo = fminf(x, y);
  y = fmaxf(x, y);
  x = lo;
}

// Move min of a[0..K-1] to a[0] and max to a[K-1], preserving the multiset.
// Pairwise round + two fold chains: ~3K/2 CEs (matches the 3K/2-2 bound).
template <int K>
__device__ __forceinline__ void minmaxK(float* a) {
  constexpr int M = K & ~1;
#pragma unroll
  for (int i = 0; i + 1 < M; i += 2) ce(a[i], a[i + 1]);   // mins->even, maxes->odd
  if constexpr (K & 1) ce(a[0], a[K - 1]);                 // fold unpaired tail
#pragma unroll
  for (int j = 2; j < M; j += 2) ce(a[0], a[j]);           // global min -> a[0]
#pragma unroll
  for (int j = 1; j < K - 1; j += 2) ce(a[j], a[K - 1]);   // global max -> a[K-1]
}

// Rounds 0..9: stream the 10 remaining SHARED values (row-major over the
// 6x6 shared core at rows 1..6, cols 1..6 of the 8x8 quad span).
template <int R>
__device__ __forceinline__ void quad_shared_rounds(float* buf, const float* w) {
  minmaxK<26 - R>(buf);
  constexpr int s = 26 + R;  // shared stream index 26..35
  buf[0] = w[(1 + s / 6) * PITCH + 1 + (s % 6)];
  if constexpr (R + 1 < 10) quad_shared_rounds<R + 1>(buf, w);
}

// Rounds 10..22: stream pixel (DX,DY)'s private 13 values:
//   R=10..16: edge row (DY?7:0), cols DX..DX+6
//   R=17..22: edge col (DX?7:0), rows 1..6
template <int R, int DX, int DY>
__device__ __forceinline__ void priv_rounds(float* buf, const float* w) {
  minmaxK<26 - R>(buf);
  if constexpr (R < 17) {
    constexpr int j = R - 10;  // 0..6
    buf[0] = w[(DY ? 7 : 0) * PITCH + DX + j];
  } else {
    constexpr int r = R - 16;  // 1..6
    buf[0] = w[r * PITCH + (DX ? 7 : 0)];
  }
  if constexpr (R + 1 < 23) priv_rounds<R + 1, DX, DY>(buf, w);
}

__device__ __forceinline__ float med3_clamp(float* buf) {
  ce(buf[0], buf[1]);
  ce(buf[1], buf[2]);
  ce(buf[0], buf[1]);
  return fminf(fmaxf(buf[1], 0.0f), 1.0f);
}

template <int DX, int DY>
__device__ __forceinline__ float pixel_median(const float keep[16], const float* w) {
  float buf[26];
#pragma unroll
  for (int j = 0; j < 16; ++j) buf[j] = keep[j];
  priv_rounds<10, DX, DY>(buf, w);  // buffer 16 -> 3
  return med3_clamp(buf);
}

extern "C" __global__ __launch_bounds__(NTHREADS)
void median7x7_hardtanh_kernel(const float* __restrict__ x, float* __restrict__ out) {
  __shared__ float tile[TROWS * PITCH];

  const int tid = threadIdx.y * 32 + threadIdx.x;
  const int tx0 = blockIdx.x * TILE_W;
  const int ty0 = blockIdx.y * TILE_H;
  const float* src = x + (size_t)blockIdx.z * (IMG_H * IMG_W);
  float* dst = out + (size_t)blockIdx.z * (IMG_H * IMG_W);

  // ---- Fill 70x22 halo tile (reflect padding resolved here, once) ----
  for (int idx = tid; idx < TROWS * PITCH; idx += NTHREADS) {
    const int py = idx / PITCH;
    const int px = idx - py * PITCH;
    const int gy = reflect_idx(ty0 - PADR + py, IMG_H);
    const int gx = reflect_idx(tx0 - PADR + px, IMG_W);
#if USE_ASYNC_LDS
    __builtin_amdgcn_global_load_async_to_lds_b32(
        (g_int_t*)(float*)(src + gy * IMG_W + gx),
        (l_int_t*)&tile[idx], 0, 0);
#else
    tile[idx] = src[gy * IMG_W + gx];
#endif
  }
#if USE_ASYNC_LDS
  __builtin_amdgcn_s_wait_asynccnt(0);
#endif
  __syncthreads();

  // ---- Each thread: one 2x2 output quad ----
  // Quad span: tile rows 2ty..2ty+7, cols 2tx..2tx+7 (8x8).
  const int qx = 2 * threadIdx.x;
  const int qy = 2 * threadIdx.y;
  const float* w = &tile[qy * PITCH + qx];

  // Seed with the first 26 of the 36 shared-core values (rows 1..6, cols 1..6).
  float buf[26];
#pragma unroll
  for (int j = 0; j < 26; ++j) buf[j] = w[(1 + j / 6) * PITCH + 1 + (j % 6)];

  quad_shared_rounds<0>(buf, w);  // stream shared 26..35; buffer 26 -> 16

  float keep[16];
#pragma unroll
  for (int j = 0; j < 16; ++j) keep[j] = buf[j];

  const float m00 = pixel_median<0, 0>(keep, w);
  const float m10 = pixel_median<1, 0>(keep, w);
  const float m01 = pixel_median<0, 1>(keep, w);
  const float m11 = pixel_median<1, 1>(keep, w);

  const int oy = ty0 + qy;
  const int ox = tx0 + qx;  // even -> 8B aligned
  *(float2*)(dst + (size_t)oy * IMG_W + ox) = make_float2(m00, m10);
  *(float2*)(dst + (size_t)(oy + 1) * IMG_W + ox) = make_float2(m01, m11);
}

extern "C" void kernel_launch(void* const* d_in, const int* in_sizes, int n_in,
                              void* d_out, int out_size, void* d_ws, size_t ws_size,
                              hipStream_t stream) {
  (void)n_in; (void)out_size; (void)d_ws; (void)ws_size;
  const float* xin = (const float*)d_in[0];
  float* out = (float*)d_out;
  const int nimg = in_sizes[0] / (IMG_H * IMG_W);  // B*C = 24
  dim3 grid(IMG_W / TILE_W, IMG_H / TILE_H, nimg);
  dim3 block(32, 8, 1);
  hipLaunchKernelGGL(median7x7_hardtanh_kernel, grid, block, 0, stream, xin, out);
}